// SilkNNUE_71038759076224
// MI455X (gfx1250) — compile-verified
//
#include <hip/hip_runtime.h>

typedef float v2f __attribute__((ext_vector_type(2)));
typedef float v8f __attribute__((ext_vector_type(8)));

#define H1_STRIDE 132   // 128 + 4 pad (floats) -> bank-conflict-free WMMA A loads
#define G3_STRIDE 68    // 64 + 4 pad
#define W2_STRIDE 132
#define W3_STRIDE 68

__global__ __launch_bounds__(128)
void nnue_kernel(const int* __restrict__ x,
                 const float* __restrict__ emb,
                 const float* __restrict__ W2,
                 const float* __restrict__ b2,
                 const float* __restrict__ W3,
                 const float* __restrict__ b3,
                 const float* __restrict__ W4,
                 float* __restrict__ out)
{
    __shared__ float s_w2[32 * W2_STRIDE];       // W2 row-major, padded stride
    __shared__ float s_w3[32 * W3_STRIDE];       // W3 row-major, padded stride
    __shared__ float s_h[4][16 * H1_STRIDE];     // per-wave activation tile (reused for CReLU tile)

    const int tid = threadIdx.x;

    // ---- stage weights into LDS (once per block) ----
    for (int e = tid; e < 32 * 128; e += 128)
        s_w2[(e >> 7) * W2_STRIDE + (e & 127)] = W2[e];
    for (int e = tid; e < 32 * 64; e += 128)
        s_w3[(e >> 6) * W3_STRIDE + (e & 63)] = W3[e];
    __syncthreads();

    const int wave = tid >> 5;
    const int lane = tid & 31;
    const int lo   = lane & 15;      // column / row-within-tile selector
    const int hi   = lane >> 4;      // half-wave selector
    const int rowbase = (blockIdx.x * 4 + wave) * 16;
    float* hbuf = s_h[wave];

    // ================= feature gather: h1 = sum of 29 embedding rows =================
    // 2 lanes per row; each lane owns 64 contiguous columns (16 x float4 accumulators).
    const int r  = lane >> 1;              // row within tile, 0..15
    const int oh = (lane & 1) * 64;        // column half offset
    const int* xrow = x + (size_t)(rowbase + r) * 32;

    float4 acc[16];
#pragma unroll
    for (int c = 0; c < 16; ++c) acc[c] = make_float4(0.f, 0.f, 0.f, 0.f);

    int id = xrow[0];
    for (int i = 0; i < 29; ++i) {
        const float4* src = (const float4*)(emb + (size_t)id * 128 + oh);
        int idn = id;
        if (i < 28) {
            idn = xrow[i + 1];
            __builtin_prefetch(emb + (size_t)idn * 128 + oh, 0, 0);
        }
#pragma unroll
        for (int c = 0; c < 16; ++c) {
            float4 v = src[c];
            acc[c].x += v.x; acc[c].y += v.y; acc[c].z += v.z; acc[c].w += v.w;
        }
        id = idn;
    }

    // ReLU + store 16x128 tile to LDS (stride 132)
#pragma unroll
    for (int c = 0; c < 16; ++c) {
        float4 v;
        v.x = fmaxf(acc[c].x, 0.f);
        v.y = fmaxf(acc[c].y, 0.f);
        v.z = fmaxf(acc[c].z, 0.f);
        v.w = fmaxf(acc[c].w, 0.f);
        *(float4*)&hbuf[r * H1_STRIDE + oh + 4 * c] = v;
    }
    // same-wave LDS is in-order; fence the compiler + drain DS before cross-lane reads
    asm volatile("s_wait_dscnt 0" ::: "memory");

    // ================= layer 2: h2 = h1 @ W2^T + b2 via V_WMMA_F32_16X16X4_F32 =================
    const int koff = 2 * hi;
    const float bb0 = b2[lo], bb1 = b2[lo + 16];
    v8f c0, c1;
#pragma unroll
    for (int j = 0; j < 8; ++j) { c0[j] = bb0; c1[j] = bb1; }

#pragma unroll
    for (int s = 0; s < 32; ++s) {
        v2f a  = *(const v2f*)&hbuf[lo * H1_STRIDE + 4 * s + koff];
        v2f bA = *(const v2f*)&s_w2[ lo       * W2_STRIDE + 4 * s + koff];
        v2f bB = *(const v2f*)&s_w2[(lo + 16) * W2_STRIDE + 4 * s + koff];
        c0 = __builtin_amdgcn_wmma_f32_16x16x4_f32(false, a, false, bA, (short)0, c0, false, false);
        c1 = __builtin_amdgcn_wmma_f32_16x16x4_f32(false, a, false, bB, (short)0, c1, false, false);
    }

    // CReLU: g3[m][0:32]=relu(h2), g3[m][32:64]=relu(-h2); reuse hbuf at stride 68
#pragma unroll
    for (int j = 0; j < 8; ++j) {
        const int m = j + 8 * hi;
        const float v0 = c0[j], v1 = c1[j];
        hbuf[m * G3_STRIDE + lo]      = fmaxf(v0, 0.f);
        hbuf[m * G3_STRIDE + lo + 32] = fmaxf(-v0, 0.f);
        hbuf[m * G3_STRIDE + lo + 16] = fmaxf(v1, 0.f);
        hbuf[m * G3_STRIDE + lo + 48] = fmaxf(-v1, 0.f);
    }
    asm volatile("s_wait_dscnt 0" ::: "memory");

    // ================= layer 3: h3 = g3 @ W3^T + b3 via WMMA =================
    const float cb0 = b3[lo], cb1 = b3[lo + 16];
    v8f d0, d1;
#pragma unroll
    for (int j = 0; j < 8; ++j) { d0[j] = cb0; d1[j] = cb1; }

#pragma unroll
    for (int s = 0; s < 16; ++s) {
        v2f a  = *(const v2f*)&hbuf[lo * G3_STRIDE + 4 * s + koff];
        v2f bA = *(const v2f*)&s_w3[ lo       * W3_STRIDE + 4 * s + koff];
        v2f bB = *(const v2f*)&s_w3[(lo + 16) * W3_STRIDE + 4 * s + koff];
        d0 = __builtin_amdgcn_wmma_f32_16x16x4_f32(false, a, false, bA, (short)0, d0, false, false);
        d1 = __builtin_amdgcn_wmma_f32_16x16x4_f32(false, a, false, bB, (short)0, d1, false, false);
    }

    // ================= layer 4: out = CReLU(h3) . W4 (VALU + half-wave reduction) ===========
    const float w4a = W4[lo];        // n = lo        (tile0, positive part)
    const float w4b = W4[lo + 32];   // n = lo + 32   (tile0, negative part)
    const float w4c = W4[lo + 16];   // n = 16 + lo   (tile1, positive part)
    const float w4d = W4[lo + 48];   // n = 48 + lo   (tile1, negative part)

    float p[8];
#pragma unroll
    for (int j = 0; j < 8; ++j) {
        const float v0 = d0[j], v1 = d1[j];
        p[j] = fmaxf(v0, 0.f) * w4a + fmaxf(-v0, 0.f) * w4b
             + fmaxf(v1, 0.f) * w4c + fmaxf(-v1, 0.f) * w4d;
    }
#pragma unroll
    for (int j = 0; j < 8; ++j) {
        p[j] += __shfl_xor(p[j], 1, 16);
        p[j] += __shfl_xor(p[j], 2, 16);
        p[j] += __shfl_xor(p[j], 4, 16);
        p[j] += __shfl_xor(p[j], 8, 16);
    }
    if (lo == 0) {
#pragma unroll
        for (int j = 0; j < 8; ++j)
            out[rowbase + 8 * hi + j] = p[j];
    }
}

extern "C" void kernel_launch(void* const* d_in, const int* in_sizes, int n_in,
                              void* d_out, int out_size, void* d_ws, size_t ws_size,
                              hipStream_t stream) {
    const int*   x   = (const int*)  d_in[0];
    const float* emb = (const float*)d_in[1];
    const float* W2  = (const float*)d_in[2];
    const float* b2  = (const float*)d_in[3];
    const float* W3  = (const float*)d_in[4];
    const float* b3  = (const float*)d_in[5];
    const float* W4  = (const float*)d_in[6];
    float* out = (float*)d_out;

    const int N = in_sizes[0] / 32;          // 131072 rows
    dim3 grid(N / 64), block(128);           // 4 waves/block * 16 rows/wave
    hipLaunchKernelGGL(nnue_kernel, grid, block, 0, stream,
                       x, emb, W2, b2, W3, b3, W4, out);
}